// SPLINE_COUPLING_LAYER_5703716569556
// MI455X (gfx1250) — compile-verified
//
#include <hip/hip_runtime.h>
#include <hip/hip_bf16.h>
#include <math.h>

// ---------------------------------------------------------------------------
// RQS coupling layer, fused (shift -> spline -> unshift) x 2 transforms.
// Memory-bound streaming kernel (~408 MB traffic -> ~17.5us @ 23.3 TB/s).
// CDNA5 path: TDM tensor_load_to_lds stages the w/h/s tables for BOTH
// transforms into LDS; transform-1 loads overlap transform-0 compute;
// s_wait_tensorcnt gates the stages. No matmul structure -> no WMMA.
// ---------------------------------------------------------------------------

#define ROWS_PER_BLOCK 16
#define WH_LDS_STRIDE  132   // 128 data floats + 4 pad floats (TDM pad feature)
#define S_ROW_FLOATS   112   // 16 dims * 7 slopes, stride 7 is bank-conflict free

typedef __attribute__((ext_vector_type(4))) unsigned int tdm_v4u;
typedef __attribute__((ext_vector_type(8))) int          tdm_v8i;
typedef __attribute__((ext_vector_type(4))) int          tdm_v4i;

#if __has_builtin(__builtin_amdgcn_tensor_load_to_lds)
#define HAVE_TDM 1
#else
#define HAVE_TDM 0
#endif

#if HAVE_TDM
// 1-D contiguous TDM copy: global -> LDS, nElem f32 elements, optional LDS
// padding (padCfg packs pad_enable/pad_interval/pad_amount bits of group1[0]).
__device__ __forceinline__ void tdm_load_1d(unsigned lds_off, const void* gptr,
                                            unsigned nElem, unsigned padCfg) {
  unsigned long long ga = (unsigned long long)(uintptr_t)gptr;
  tdm_v4u g0;
  g0.x = 1u;                                            // count=1 (valid), user mode
  g0.y = lds_off;                                       // LDS byte address
  g0.z = (unsigned)(ga & 0xFFFFFFFFull);                // global_addr[31:0]
  g0.w = (unsigned)((ga >> 32) & 0x1FFFFFFull)          // global_addr[56:32]
       | (2u << 30);                                    // type = 2 ("image")
  tdm_v8i g1;
  g1[0] = (int)((2u << 16) | padCfg);                   // data_size=4B | pad cfg
  g1[1] = (int)((nElem & 0xFFFFu) << 16);               // tensor_dim0[15:0]
  g1[2] = (int)((nElem >> 16) | (1u << 16));            // tensor_dim0[31:16] | tensor_dim1=1
  g1[3] = (int)((nElem & 0xFFFFu) << 16);               // tile_dim0 = nElem
  g1[4] = 1;                                            // tile_dim1=1, tile_dim2=0
  g1[5] = (int)nElem;                                   // tensor_dim0_stride lo
  g1[6] = 0;
  g1[7] = 0;
  tdm_v4i z4 = (tdm_v4i)0;
#if defined(__clang_major__) && (__clang_major__ >= 23)
  tdm_v8i z8 = (tdm_v8i)0;
  __builtin_amdgcn_tensor_load_to_lds(g0, g1, z4, z4, z8, 0);
#else
  __builtin_amdgcn_tensor_load_to_lds(g0, g1, z4, z4, 0);
#endif
}
#endif  // HAVE_TDM

__device__ __forceinline__ float frcp(float x) { return __builtin_amdgcn_rcpf(x); }

// One transform: periodic shift -> RQS forward -> inverse shift.
// wv/hv point at this element's 8 contiguous logits in LDS, sv at its 7 slopes.
__device__ __forceinline__ void rqs_step(const float* wv, const float* hv,
                                         const float* sv, float sh,
                                         float& x, float& lsum) {
  const float c = 1.992f;  // (XY_MAX-XY_MIN) - N_BINS*MIN_BW

  // shift forward: mod(x + sh + 1, 2) - 1
  float z = x + sh + 1.0f;
  z = z - 2.0f * floorf(z * 0.5f);
  z = z - 1.0f;

  float wu[8], hu[8];
  {
    const float4* p = reinterpret_cast<const float4*>(wv);
    float4 a = p[0], b = p[1];
    wu[0]=a.x; wu[1]=a.y; wu[2]=a.z; wu[3]=a.w;
    wu[4]=b.x; wu[5]=b.y; wu[6]=b.z; wu[7]=b.w;
    const float4* q = reinterpret_cast<const float4*>(hv);
    float4 e = q[0], f = q[1];
    hu[0]=e.x; hu[1]=e.y; hu[2]=e.z; hu[3]=e.w;
    hu[4]=f.x; hu[5]=f.y; hu[6]=f.z; hu[7]=f.w;
  }

  // softmax(w)*c + MIN_BW ; softmax(h)*c + MIN_BW
  float mw = wu[0], mh = hu[0];
#pragma unroll
  for (int k = 1; k < 8; ++k) { mw = fmaxf(mw, wu[k]); mh = fmaxf(mh, hu[k]); }
  float ew[8], eh[8], sw = 0.f, shsum = 0.f;
#pragma unroll
  for (int k = 0; k < 8; ++k) {
    ew[k] = __expf(wu[k] - mw); sw += ew[k];
    eh[k] = __expf(hu[k] - mh); shsum += eh[k];
  }
  float rw = frcp(sw) * c, rh = frcp(shsum) * c;
  float bw[8], bh[8];
#pragma unroll
  for (int k = 0; k < 8; ++k) {
    bw[k] = ew[k] * rw + 0.001f;
    bh[k] = eh[k] * rh + 0.001f;
  }

  // knot slopes: d = [1, softplus(s)+MIN_KS (x7), 1]
  float d[9];
  d[0] = 1.0f; d[8] = 1.0f;
#pragma unroll
  for (int k = 0; k < 7; ++k) {
    float v = sv[k];
    d[k + 1] = fmaxf(v, 0.0f) + __logf(1.0f + __expf(-fabsf(v))) + 0.001f;
  }

  // bin search over interior knots (cumsum on the fly)
  int idx = 0;
  float x0 = -1.0f, y0 = -1.0f, cw = -1.0f, ch = -1.0f;
#pragma unroll
  for (int k = 0; k < 7; ++k) {
    cw += bw[k]; ch += bh[k];
    if (z >= cw) { idx = k + 1; x0 = cw; y0 = ch; }
  }
  float wk = bw[0], hk = bh[0], d0 = d[0], d1 = d[1];
#pragma unroll
  for (int k = 1; k < 8; ++k) {
    bool sel = (idx == k);
    wk = sel ? bw[k] : wk;
    hk = sel ? bh[k] : hk;
    d0 = sel ? d[k]  : d0;
    d1 = sel ? d[k + 1] : d1;
  }

  // rational quadratic spline
  float rwk = frcp(wk);
  float sk  = hk * rwk;
  float tt  = (z - x0) * rwk;
  float omt = 1.0f - tt;
  float tom = tt * omt;
  float denom = sk + (d1 + d0 - 2.0f * sk) * tom;
  float y = y0 + hk * (sk * tt * tt + d0 * tom) * frcp(denom);
  float num = d1 * tt * tt + 2.0f * sk * tom + d0 * omt * omt;
  lsum += 2.0f * __logf(sk) + __logf(num) - 2.0f * __logf(denom);

  // inverse shift: mod(y - sh + 1, 2) - 1
  float u = y - sh + 1.0f;
  u = u - 2.0f * floorf(u * 0.5f);
  x = u - 1.0f;
}

__global__ __launch_bounds__(256)
void rqs_tdm_kernel(const float* __restrict__ X,  const float* __restrict__ W,
                    const float* __restrict__ H,  const float* __restrict__ S,
                    const float* __restrict__ SH, float* __restrict__ out, int m) {
  __shared__ float wbuf[2][ROWS_PER_BLOCK * WH_LDS_STRIDE];
  __shared__ float hbuf[2][ROWS_PER_BLOCK * WH_LDS_STRIDE];
  __shared__ float sbuf[2][ROWS_PER_BLOCK * S_ROW_FLOATS];

  const int row0 = blockIdx.x * ROWS_PER_BLOCK;
  const int rows = (m - row0 < ROWS_PER_BLOCK) ? (m - row0) : ROWS_PER_BLOCK;
  const int tid  = threadIdx.x;

#if HAVE_TDM
  // Wave 0 issues all 6 TDM descriptors (both transforms) back-to-back so
  // transform-1 tables stream in while transform 0 computes.
  if (tid < 32) {
    // pad_enable | pad_interval=6 (every 128 DWORDs) | pad_amount=3 (4 DWORDs)
    const unsigned padWH = (1u << 20) | (6u << 22) | (3u << 25);
#pragma unroll
    for (int t = 0; t < 2; ++t) {
      const size_t base = (size_t)t * (size_t)m + (size_t)row0;
      tdm_load_1d((unsigned)(uintptr_t)&wbuf[t][0], W + base * 128,
                  (unsigned)(rows * 128), padWH);
      tdm_load_1d((unsigned)(uintptr_t)&hbuf[t][0], H + base * 128,
                  (unsigned)(rows * 128), padWH);
      tdm_load_1d((unsigned)(uintptr_t)&sbuf[t][0], S + base * S_ROW_FLOATS,
                  (unsigned)(rows * S_ROW_FLOATS), 0u);
    }
    __builtin_amdgcn_s_wait_tensorcnt(3);  // in-order: transform-0 tables done
  }
#else
  // Fallback: cooperative vector copy (padded stride emulated for w/h).
  for (int t = 0; t < 2; ++t) {
    const size_t base = (size_t)t * (size_t)m + (size_t)row0;
    for (int i = tid; i < rows * 128; i += 256) {
      int r = i >> 7, cidx = i & 127;
      wbuf[t][r * WH_LDS_STRIDE + cidx] = W[base * 128 + i];
      hbuf[t][r * WH_LDS_STRIDE + cidx] = H[base * 128 + i];
    }
    for (int i = tid; i < rows * S_ROW_FLOATS; i += 256)
      sbuf[t][i] = S[base * S_ROW_FLOATS + i];
  }
#endif
  __syncthreads();

  const int rl = tid >> 4;     // local row
  const int dm = tid & 15;     // dim
  const bool active = rl < rows;
  const size_t gelem = (size_t)row0 * 16 + (size_t)tid;

  float x = 0.0f, lsum = 0.0f;
  if (active) {
    x = X[gelem];
    float sh0 = SH[gelem];  // transform 0 shifts
    rqs_step(&wbuf[0][rl * WH_LDS_STRIDE + dm * 8],
             &hbuf[0][rl * WH_LDS_STRIDE + dm * 8],
             &sbuf[0][rl * S_ROW_FLOATS + dm * 7], sh0, x, lsum);
  }
#if HAVE_TDM
  if (tid < 32) __builtin_amdgcn_s_wait_tensorcnt(0);
#endif
  __syncthreads();
  if (active) {
    float sh1 = SH[(size_t)m * 16 + gelem];  // transform 1 shifts
    rqs_step(&wbuf[1][rl * WH_LDS_STRIDE + dm * 8],
             &hbuf[1][rl * WH_LDS_STRIDE + dm * 8],
             &sbuf[1][rl * S_ROW_FLOATS + dm * 7], sh1, x, lsum);
    out[gelem] = x;
    out[(size_t)m * 16 + gelem] = lsum;
  }
}

extern "C" void kernel_launch(void* const* d_in, const int* in_sizes, int n_in,
                              void* d_out, int out_size, void* d_ws, size_t ws_size,
                              hipStream_t stream) {
  const float* X  = (const float*)d_in[0];
  const float* W  = (const float*)d_in[1];
  const float* H  = (const float*)d_in[2];
  const float* S  = (const float*)d_in[3];
  const float* SH = (const float*)d_in[4];
  float* out = (float*)d_out;
  const int m = in_sizes[0] / 16;
  const int grid = (m + ROWS_PER_BLOCK - 1) / ROWS_PER_BLOCK;
  rqs_tdm_kernel<<<grid, 256, 0, stream>>>(X, W, H, S, SH, out, m);
}